// _NonLocalBlockND_30562987278417
// MI455X (gfx1250) — compile-verified
//
#include <hip/hip_runtime.h>
#include <math.h>

#define IC   8
#define CIN  16
#define HMS  96          // x_ms spatial
#define HC   48          // coarse grid
#define NC   (HC*HC)     // 2304 coarse positions
#define HP   192         // pan spatial

typedef __attribute__((ext_vector_type(2))) float v2f;
typedef __attribute__((ext_vector_type(8))) float v8f;

// ---------------- K1: theta = conv3x3 s2 p1 on x_ms -> th[NC][IC] ----------------
__global__ __launch_bounds__(256) void theta_kernel(
    const float* __restrict__ x, const float* __restrict__ w,
    const float* __restrict__ b, float* __restrict__ th)
{
  int gid = blockIdx.x * 256 + threadIdx.x;
  if (gid >= NC * IC) return;
  int oc = gid & 7, pos = gid >> 3;
  int oy = pos / HC, ox = pos % HC;
  float acc = b[oc];
  for (int ic = 0; ic < CIN; ++ic) {
    const float* xp = x + ic * HMS * HMS;
    const float* wp = w + (oc * CIN + ic) * 9;
#pragma unroll
    for (int ky = 0; ky < 3; ++ky) {
      int iy = 2 * oy - 1 + ky;
      if ((unsigned)iy >= HMS) continue;
#pragma unroll
      for (int kx = 0; kx < 3; ++kx) {
        int ix = 2 * ox - 1 + kx;
        if ((unsigned)ix >= HMS) continue;
        acc += wp[ky * 3 + kx] * xp[iy * HMS + ix];
      }
    }
  }
  th[pos * IC + oc] = acc;
}

// -------- K2: phi = maxpool2(conv3x3 s2 p1 on x_pan) -> ph[NC][IC] (coarse) --------
__global__ __launch_bounds__(256) void phi_kernel(
    const float* __restrict__ x, const float* __restrict__ w,
    const float* __restrict__ b, float* __restrict__ ph)
{
  int gid = blockIdx.x * 256 + threadIdx.x;
  if (gid >= NC * IC) return;
  int oc = gid & 7, pos = gid >> 3;
  int cy = pos / HC, cx = pos % HC;
  float m = -INFINITY;
#pragma unroll
  for (int dy = 0; dy < 2; ++dy)
#pragma unroll
    for (int dx = 0; dx < 2; ++dx) {
      int fy = 2 * cy + dy, fx = 2 * cx + dx;   // conv-output (96x96) coords
      float acc = b[oc];
      for (int ic = 0; ic < CIN; ++ic) {
        const float* xp = x + ic * HP * HP;
        const float* wp = w + (oc * CIN + ic) * 9;
#pragma unroll
        for (int ky = 0; ky < 3; ++ky) {
          int iy = 2 * fy - 1 + ky;
          if ((unsigned)iy >= HP) continue;
#pragma unroll
          for (int kx = 0; kx < 3; ++kx) {
            int ix = 2 * fx - 1 + kx;
            if ((unsigned)ix >= HP) continue;
            acc += wp[ky * 3 + kx] * xp[iy * HP + ix];
          }
        }
      }
      m = fmaxf(m, acc);
    }
  ph[pos * IC + oc] = m;
}

// ---- K3: g fine = maxpool2(conv1x1(hp_pan)); per coarse block: sum-of-4 + top-left ----
__global__ __launch_bounds__(256) void g_kernel(
    const float* __restrict__ hp, const float* __restrict__ w,
    const float* __restrict__ b, float* __restrict__ g4,
    float* __restrict__ gtl)
{
  int gid = blockIdx.x * 256 + threadIdx.x;
  if (gid >= NC * IC) return;
  int oc = gid & 7, pos = gid >> 3;
  int cy = pos / HC, cx = pos % HC;
  float sum = 0.f, tl = 0.f;
#pragma unroll
  for (int dy = 0; dy < 2; ++dy)
#pragma unroll
    for (int dx = 0; dx < 2; ++dx) {
      int fy = 2 * cy + dy, fx = 2 * cx + dx;   // fine g coords (96x96)
      float m = -INFINITY;
#pragma unroll
      for (int py = 0; py < 2; ++py)
#pragma unroll
        for (int px = 0; px < 2; ++px) {
          int iy = 2 * fy + py, ix = 2 * fx + px;  // 192x192 coords
          float acc = b[oc];
          for (int ic = 0; ic < CIN; ++ic)
            acc += w[oc * CIN + ic] * hp[ic * HP * HP + iy * HP + ix];
          m = fmaxf(m, acc);
        }
      sum += m;
      if (dy == 0 && dx == 0) tl = m;
    }
  g4[pos * IC + oc]  = sum;
  gtl[pos * IC + oc] = tl;
}

// ---------------- monotone float<->u32 sort-key helpers ----------------
__device__ __forceinline__ unsigned fkey(float f) {
  unsigned u = __float_as_uint(f);
  return u ^ (unsigned)(((int)u >> 31) | 0x80000000);
}
__device__ __forceinline__ float fdecode(unsigned u) {
  // inverse of fkey, branchless: top bit set -> was positive
  unsigned mask = ((unsigned)(~((int)u >> 31))) | 0x80000000u;
  return __uint_as_float(u ^ mask);
}
__device__ __forceinline__ unsigned long long umax64(unsigned long long a,
                                                     unsigned long long b) {
  return a > b ? a : b;
}
__device__ __forceinline__ unsigned long long umin64(unsigned long long a,
                                                     unsigned long long b) {
  return a < b ? a : b;
}

// ---- K4: WMMA f32 similarity + fused per-row top-2 + softmax-weighted gather ----
// One wave per 16-row tile. D-tile layout: VGPR v, lane l -> row v+8*(l>=16), col l%16.
// top-2 tracked as 64-bit keys (monotone(value) << 32 | ~col): pure min/max, no branches.
__global__ __launch_bounds__(32) void attn_topk_kernel(
    const float* __restrict__ th,   // [NC][IC]  A rows
    const float* __restrict__ ph,   // [NC][IC]  B cols (B[k][n] = ph[n][k])
    const float* __restrict__ g4,   // [NC][IC]  sum of 4 fine g per block
    const float* __restrict__ gtl,  // [NC][IC]  top-left fine g per block
    float* __restrict__ y)          // [NC][IC]
{
  const int lane = threadIdx.x;
  const int half = lane >> 4;
  const int lp   = lane & 15;
  const int i0   = blockIdx.x * 16;

  // A operands (loop-invariant): lane l holds A[l%16][k], k = 2*half (+1), and +4 for 2nd WMMA
  const int row = i0 + lp;
  const int ka  = half * 2;
  v2f a0 = *(const v2f*)(th + row * IC + ka);
  v2f a1 = *(const v2f*)(th + row * IC + 4 + ka);

  unsigned long long k1[8], k2[8];
#pragma unroll
  for (int v = 0; v < 8; ++v) { k1[v] = 0ull; k2[v] = 0ull; }

  for (int jt = 0; jt < NC / 16; ++jt) {
    const int col = jt * 16 + lp;
    v2f b0 = *(const v2f*)(ph + col * IC + ka);
    v2f b1 = *(const v2f*)(ph + col * IC + 4 + ka);
    __builtin_prefetch(ph + (col + 16) * IC, 0, 0);   // global_prefetch_b8

    v8f c = {};
    c = __builtin_amdgcn_wmma_f32_16x16x4_f32(false, a0, false, b0, (short)0, c, false, false);
    c = __builtin_amdgcn_wmma_f32_16x16x4_f32(false, a1, false, b1, (short)0, c, false, false);

    const unsigned invcol = ~(unsigned)col;
#pragma unroll
    for (int v = 0; v < 8; ++v) {
      unsigned long long k =
          ((unsigned long long)fkey(c[v]) << 32) | invcol;
      unsigned long long hi = umax64(k, k1[v]);
      unsigned long long lo = umin64(k, k1[v]);
      k1[v] = hi;
      k2[v] = umax64(k2[v], lo);
    }
  }

  // merge top-2 across the 16 lanes of each half-wave (xor butterfly stays in-half)
#pragma unroll
  for (int v = 0; v < 8; ++v) {
    for (int off = 1; off < 16; off <<= 1) {
      unsigned long long o1 = __shfl_xor(k1[v], off, 32);
      unsigned long long o2 = __shfl_xor(k2[v], off, 32);
      unsigned long long hi = umax64(k1[v], o1);
      unsigned long long lo = umin64(k1[v], o1);
      k1[v] = hi;
      k2[v] = umax64(lo, umax64(k2[v], o2));
    }
  }

  // softmax over [V1,V1,V1,V1,V2]: weights [1,1,1,1,e]/(4+e), e = exp(V2-V1)
#pragma unroll
  for (int v = 0; v < 8; ++v) {
    const int i = i0 + half * 8 + v;
    float v1 = fdecode((unsigned)(k1[v] >> 32));
    float v2 = fdecode((unsigned)(k2[v] >> 32));
    int   j1 = (int)(~(unsigned)k1[v] & 0xFFFFFFFFu);
    int   j2 = (int)(~(unsigned)k2[v] & 0xFFFFFFFFu);
    float e   = __expf(v2 - v1);
    float inv = 1.0f / (4.0f + e);
    if (lp < IC) {
      float gs = g4[j1 * IC + lp];
      float gt = gtl[j2 * IC + lp];
      y[i * IC + lp] = (gs + e * gt) * inv;
    }
  }
}

// ---- K5: 1x1 conv W + train-mode BN (stats over coarse == fine) + 2x upsample ----
__global__ __launch_bounds__(256) void wbn_kernel(
    const float* __restrict__ yc,  // [NC][IC]
    const float* __restrict__ Ww,  // [CIN][IC]
    const float* __restrict__ Wb,
    const float* __restrict__ gamma,
    const float* __restrict__ beta,
    float* __restrict__ out)       // [CIN][96][96]
{
  __shared__ float ly[NC];
  __shared__ float red[256], red2[256];
  const int ch = blockIdx.x;
  const int t  = threadIdx.x;
  float w[IC];
#pragma unroll
  for (int c = 0; c < IC; ++c) w[c] = Ww[ch * IC + c];
  const float bias = Wb[ch];
  float s = 0.f, s2 = 0.f;
  for (int i = t; i < NC; i += 256) {
    float acc = bias;
#pragma unroll
    for (int c = 0; c < IC; ++c) acc += w[c] * yc[i * IC + c];
    ly[i] = acc; s += acc; s2 += acc * acc;
  }
  red[t] = s; red2[t] = s2;
  __syncthreads();
  for (int off = 128; off > 0; off >>= 1) {
    if (t < off) { red[t] += red[t + off]; red2[t] += red2[t + off]; }
    __syncthreads();
  }
  float mean  = red[0] * (1.0f / NC);
  float var   = red2[0] * (1.0f / NC) - mean * mean;
  float scale = gamma[ch] * rsqrtf(var + 1e-5f);
  float shift = beta[ch] - mean * scale;
  for (int p = t; p < HMS * HMS; p += 256) {
    int yy = p / HMS, xx = p % HMS;
    out[ch * HMS * HMS + p] = ly[(yy >> 1) * HC + (xx >> 1)] * scale + shift;
  }
}

extern "C" void kernel_launch(void* const* d_in, const int* in_sizes, int n_in,
                              void* d_out, int out_size, void* d_ws, size_t ws_size,
                              hipStream_t stream)
{
  const float* x_ms   = (const float*)d_in[0];
  const float* hp_pan = (const float*)d_in[1];
  const float* x_pan  = (const float*)d_in[2];
  const float* g_w    = (const float*)d_in[3];
  const float* g_b    = (const float*)d_in[4];
  const float* t_w    = (const float*)d_in[5];
  const float* t_b    = (const float*)d_in[6];
  const float* p_w    = (const float*)d_in[7];
  const float* p_b    = (const float*)d_in[8];
  const float* W_w    = (const float*)d_in[9];
  const float* W_b    = (const float*)d_in[10];
  const float* gamma  = (const float*)d_in[11];
  const float* beta   = (const float*)d_in[12];
  float* out = (float*)d_out;

  char* ws = (char*)d_ws;
  const size_t SZ = (size_t)NC * IC * sizeof(float);
  float* th  = (float*)(ws);
  float* ph  = (float*)(ws + SZ);
  float* g4  = (float*)(ws + 2 * SZ);
  float* gtl = (float*)(ws + 3 * SZ);
  float* yc  = (float*)(ws + 4 * SZ);

  const int nb = (NC * IC + 255) / 256;   // 72
  theta_kernel<<<nb, 256, 0, stream>>>(x_ms, t_w, t_b, th);
  phi_kernel  <<<nb, 256, 0, stream>>>(x_pan, p_w, p_b, ph);
  g_kernel    <<<nb, 256, 0, stream>>>(hp_pan, g_w, g_b, g4, gtl);
  attn_topk_kernel<<<NC / 16, 32, 0, stream>>>(th, ph, g4, gtl, yc);
  wbn_kernel  <<<CIN, 256, 0, stream>>>(yc, W_w, W_b, gamma, beta, out);
  // second tuple element: x_pan passthrough
  hipMemcpyAsync(out + CIN * HMS * HMS, x_pan,
                 (size_t)CIN * HP * HP * sizeof(float),
                 hipMemcpyDeviceToDevice, stream);
}